// GaussianYOLOLayer_57526791963199
// MI455X (gfx1250) — compile-verified
//
#include <hip/hip_runtime.h>
#include <cstdint>
#include <cstddef>

// ---------------------------------------------------------------------------
// YOLO head decode: [B, 258, 76, 76] -> [B, 3*76*76, 86]
// Bandwidth-bound transpose + activation. LDS-staged, async global->LDS on
// CDNA5 (ASYNCcnt path), coalesced b128 loads / b32 stores.
// ---------------------------------------------------------------------------

#define GDIM    76
#define GG      (GDIM * GDIM)      // 5776
#define NA      3
#define CH      86                 // channels per anchor (nC + 6)
#define CTOT    (NA * CH)          // 258
#define TILE    128                // spatial positions per block
#define NT      46                 // ceil(5776 / 128)
#define LDS_W   132                // padded row stride (floats): 528B, 16B-aligned, 2-way bank conflict max
#define BLK     256
#define STRIDE_F 8.0f              // 608 / 76

#if defined(__has_builtin)
#  if __has_builtin(__builtin_amdgcn_global_load_async_to_lds_b128)
#    define HAVE_ASYNC_LDS 1
#  endif
#  if __has_builtin(__builtin_amdgcn_s_wait_asynccnt)
#    define HAVE_WAIT_ASYNC 1
#  endif
#endif

#ifdef HAVE_ASYNC_LDS
typedef int v4i __attribute__((ext_vector_type(4)));
typedef __attribute__((address_space(1))) v4i GV4;   // global-AS int4
typedef __attribute__((address_space(3))) v4i LV4;   // LDS-AS int4
#endif

__device__ __forceinline__ float fast_sigmoid(float x) {
  // 1 / (1 + e^-x): native v_exp_f32 + v_rcp_f32
  return __builtin_amdgcn_rcpf(1.0f + __expf(-x));
}

__global__ __launch_bounds__(BLK)
void yolo_head_decode(const float* __restrict__ x, float* __restrict__ out, int B) {
  __shared__ __attribute__((aligned(16))) float smem[CH * LDS_W];  // 45,408 B

  const int t  = blockIdx.x % NT;
  const int a  = (blockIdx.x / NT) % NA;
  const int b  = blockIdx.x / (NT * NA);
  const int g0 = t * TILE;
  const int validP = (GG - g0) < TILE ? (GG - g0) : TILE;   // last tile = 16
  const int tid = threadIdx.x;

  // ---------------- stage 1: global (channel-planar) -> LDS -----------------
  // 86 channels x 32 float4 chunks = 2752 chunks; one b128 per lane per iter.
  const float* xa = x + ((size_t)b * CTOT + (size_t)a * CH) * GG;
  const int nchunks = CH * (TILE / 4);
  for (int m = tid; m < nchunks; m += BLK) {
    const int c = m >> 5;            // chunk's channel (32 chunks per channel)
    const int q = m & 31;            // float4 index within tile
    const int g = g0 + (q << 2);
    if (g < GG) {                    // GG % 4 == 0, tiles 128-aligned: chunk all-or-nothing
      const float* src = xa + (size_t)c * GG + g;
      float*       dst = &smem[c * LDS_W + (q << 2)];
#ifdef HAVE_ASYNC_LDS
      __builtin_amdgcn_global_load_async_to_lds_b128((GV4*)src, (LV4*)dst, 0, 0);
#else
      *(float4*)dst = *(const float4*)src;
#endif
    }
  }
#ifdef HAVE_ASYNC_LDS
#  ifdef HAVE_WAIT_ASYNC
  __builtin_amdgcn_s_wait_asynccnt(0);
#  else
  asm volatile("s_wait_asynccnt 0" ::: "memory");
#  endif
#endif
  __syncthreads();

  // ---------------- stage 2: LDS -> global (location-major) -----------------
  // Flat index o = p*86 + c walked by consecutive lanes -> contiguous stores.
  const float AW = (a == 0) ? 116.0f : ((a == 1) ? 156.0f : 373.0f);
  const float AH = (a == 0) ?  90.0f : ((a == 1) ? 198.0f : 326.0f);
  float* ob = out + ((size_t)b * (NA * GG) + (size_t)a * GG + g0) * CH;

  const int nout = validP * CH;
  for (int o = tid; o < nout; o += BLK) {
    const unsigned p = (unsigned)o / (unsigned)CH;     // magic-number div
    const unsigned c = (unsigned)o - p * (unsigned)CH;
    const float v = smem[c * LDS_W + p];               // transposed read (2-way conflict max)
    const int g = g0 + (int)p;
    float r;
    if (c == 0) {
      r = (fast_sigmoid(v) + (float)(g % GDIM)) * STRIDE_F;          // x
    } else if (c == 1) {
      r = (fast_sigmoid(v) + (float)(g / GDIM)) * STRIDE_F;          // y
    } else if (c == 2) {
      r = __expf(v) * AW;   // exp(w) * (ANCHOR/stride) * stride == exp(w)*ANCHOR
    } else if (c == 3) {
      r = __expf(v) * AH;
    } else {
      r = fast_sigmoid(v);  // conf (c==4) and classes (c>=5)
    }
    ob[o] = r;
  }
}

extern "C" void kernel_launch(void* const* d_in, const int* in_sizes, int n_in,
                              void* d_out, int out_size, void* d_ws, size_t ws_size,
                              hipStream_t stream) {
  (void)n_in; (void)out_size; (void)d_ws; (void)ws_size;
  const float* x = (const float*)d_in[0];
  float* out = (float*)d_out;
  const int B = in_sizes[0] / (CTOT * GG);             // 32 for the reference shapes
  const int nblocks = B * NA * NT;
  yolo_head_decode<<<nblocks, BLK, 0, stream>>>(x, out, B);
}